// Pooler_8340826489238
// MI455X (gfx1250) — compile-verified
//
#include <hip/hip_runtime.h>
#include <math.h>

#define PH 7
#define PW 7
#define NSAMP 14          // PH * SR (= PW * SR)
#define C_CH 256
#define OUT_PER_ROI (C_CH * PH * PW)          // 12544 floats = 50176 bytes
#define CHUNKS_PER_ROI (OUT_PER_ROI / 4)      // 3136 float4 chunks

__global__ __launch_bounds__(256)
void roi_align_fpn_kernel(const float* __restrict__ x0, const float* __restrict__ x1,
                          const float* __restrict__ x2, const float* __restrict__ x3,
                          const float* __restrict__ boxes, float* __restrict__ out,
                          int R, int N)
{
    // Per-ROI sample metadata, axis 0 = y, axis 1 = x. Block-uniform, staged in LDS.
    __shared__ int   s_lo[2][NSAMP];
    __shared__ int   s_hi[2][NSAMP];
    __shared__ float s_w0[2][NSAMP];
    __shared__ float s_w1[2][NSAMP];
    // Result staging: one contiguous 49KB tile per ROI, drained by async DMA.
    // Lane write addresses stride by 49 words -> 49 mod 64 banks: conflict-free.
    __shared__ __align__(16) float s_res[OUT_PER_ROI];

    const int roi = blockIdx.x;
    const int c   = threadIdx.x;

    // Box is uniform across the block (compiler scalarizes to s_load_b128).
    const float bx1 = boxes[roi * 4 + 0];
    const float by1 = boxes[roi * 4 + 1];
    const float bx2 = boxes[roi * 4 + 2];
    const float by2 = boxes[roi * 4 + 3];

    // Level assignment: floor(4 + log2(sqrt(area)/224 + 1e-6)), clip [2,5], -2.
    const float area = (bx2 - bx1) * (by2 - by1);
    const float sz   = sqrtf(area);
    float lvlf = floorf(4.0f + log2f(sz * (1.0f / 224.0f) + 1e-6f));
    lvlf = fminf(fmaxf(lvlf, 2.0f), 5.0f);
    const int level = (int)lvlf - 2;

    const float* feat;
    int H, W;
    float scale;
    if (level == 0)      { feat = x0; H = 200; W = 200; scale = 0.25f;    }
    else if (level == 1) { feat = x1; H = 100; W = 100; scale = 0.125f;   }
    else if (level == 2) { feat = x2; H = 50;  W = 50;  scale = 0.0625f;  }
    else                 { feat = x3; H = 25;  W = 25;  scale = 0.03125f; }

    const float ry1 = by1 * scale, ry2 = by2 * scale;
    const float rx1 = bx1 * scale, rx2 = bx2 * scale;
    const float roi_h = fmaxf(ry2 - ry1, 1.0f);
    const float roi_w = fmaxf(rx2 - rx1, 1.0f);
    const float bin_h = roi_h * (1.0f / PH);
    const float bin_w = roi_w * (1.0f / PW);

    // Threads 0..27 build sample metadata; empty-predicate folded into weights
    // (empty <=> y-cond OR x-cond; zeroing either axis' weight pair zeroes the
    // sample contribution exactly, matching where(empty, 0, v)).
    if (threadIdx.x < 2 * NSAMP) {
        const int axis = threadIdx.x / NSAMP;     // 0 -> y, 1 -> x
        const int i    = threadIdx.x % NSAMP;
        const float start = axis ? rx1 : ry1;
        const float binsz = axis ? bin_w : bin_h;
        const int   dim   = axis ? W : H;
        // coord = start + (i/2)*bin + (i%2 + 0.5) * bin/2   (SR == 2)
        float coord = start + (float)(i >> 1) * binsz
                    + ((float)(i & 1) + 0.5f) * (binsz * 0.5f);
        const bool empty = (coord < -1.0f) || (coord > (float)dim);
        float cc = fmaxf(coord, 0.0f);
        const float fl = floorf(cc);
        const bool clampHi = fl >= (float)(dim - 1);
        cc = clampHi ? (float)(dim - 1) : cc;
        const int lo = clampHi ? (dim - 1) : (int)fl;
        const int hi = clampHi ? (dim - 1) : (lo + 1);
        const float l = cc - (float)lo;
        s_lo[axis][i] = lo;
        s_hi[axis][i] = hi;
        s_w0[axis][i] = empty ? 0.0f : (1.0f - l);
        s_w1[axis][i] = empty ? 0.0f : l;
    }
    __syncthreads();

    const int bidx = roi / N;
    const float* plane = feat + (size_t)(bidx * C_CH + c) * (size_t)(H * W);

    // Warm this channel's row window (gfx1250 global_prefetch_b8; no counters).
    {
        const int px0 = s_lo[1][0];
        for (int i = 0; i < NSAMP; ++i) {
            __builtin_prefetch((const void*)(plane + s_lo[0][i] * W + px0), 0, 1);
        }
    }

    float* lrow = &s_res[c * (PH * PW)];

    for (int ph = 0; ph < PH; ++ph) {
        // Two y-samples for this bin row: hoist row pointers + weights.
        const int   yAlo = s_lo[0][2 * ph],     yAhi = s_hi[0][2 * ph];
        const float wA0  = s_w0[0][2 * ph],     wA1  = s_w1[0][2 * ph];
        const int   yBlo = s_lo[0][2 * ph + 1], yBhi = s_hi[0][2 * ph + 1];
        const float wB0  = s_w0[0][2 * ph + 1], wB1  = s_w1[0][2 * ph + 1];
        const float* rA0 = plane + yAlo * W;
        const float* rA1 = plane + yAhi * W;
        const float* rB0 = plane + yBlo * W;
        const float* rB1 = plane + yBhi * W;

        #pragma unroll
        for (int pw = 0; pw < PW; ++pw) {
            float acc = 0.0f;
            #pragma unroll
            for (int sx = 0; sx < 2; ++sx) {
                const int xi = 2 * pw + sx;
                const int   xl  = s_lo[1][xi], xh = s_hi[1][xi];
                const float wx0 = s_w0[1][xi], wx1 = s_w1[1][xi];
                // y-sample A (sy = 0)
                acc += wA0 * (wx0 * rA0[xl] + wx1 * rA0[xh])
                     + wA1 * (wx0 * rA1[xl] + wx1 * rA1[xh]);
                // y-sample B (sy = 1)
                acc += wB0 * (wx0 * rB0[xl] + wx1 * rB0[xh])
                     + wB1 * (wx0 * rB1[xl] + wx1 * rB1[xh]);
            }
            lrow[ph * PW + pw] = acc * 0.25f;   // ds_store_b32, conflict-free
        }
    }

    __syncthreads();   // all LDS results visible before DMA drain

    // Drain the 49KB LDS tile to global with the CDNA5 async LDS->global DMA.
    // Lane k stores chunk k*16B: each wave emits fully coalesced 512B stores,
    // tracked on ASYNCcnt (not the VMEM store path).
    {
        const unsigned long long out_base =
            (unsigned long long)(uintptr_t)(out + (size_t)roi * OUT_PER_ROI);
        // Low 32 bits of a flat LDS address = offset within the LDS allocation.
        const unsigned lds_base = (unsigned)(uintptr_t)(&s_res[0]);
        for (int k = threadIdx.x; k < CHUNKS_PER_ROI; k += C_CH) {
            const unsigned goff = (unsigned)k * 16u;
            const unsigned loff = lds_base + (unsigned)k * 16u;
            asm volatile("global_store_async_from_lds_b128 %0, %1, %2"
                         :
                         : "v"(goff), "v"(loff), "s"(out_base)
                         : "memory");
        }
        asm volatile("s_wait_asynccnt 0x0" ::: "memory");
    }
}

extern "C" void kernel_launch(void* const* d_in, const int* in_sizes, int n_in,
                              void* d_out, int out_size, void* d_ws, size_t ws_size,
                              hipStream_t stream) {
    (void)n_in; (void)out_size; (void)d_ws; (void)ws_size;
    const float* x0    = (const float*)d_in[0];
    const float* x1    = (const float*)d_in[1];
    const float* x2    = (const float*)d_in[2];
    const float* x3    = (const float*)d_in[3];
    const float* boxes = (const float*)d_in[4];
    float* out = (float*)d_out;

    const int R = in_sizes[4] / 4;                       // total ROIs (B*N)
    const int B = in_sizes[0] / (C_CH * 200 * 200);      // batch from x0 size
    const int N = R / (B > 0 ? B : 1);                   // ROIs per batch image

    dim3 grid(R), block(C_CH);
    hipLaunchKernelGGL(roi_align_fpn_kernel, grid, block, 0, stream,
                       x0, x1, x2, x3, boxes, out, R, N);
}